// LocalGroupedZernikeNewBP_86294482911420
// MI455X (gfx1250) — compile-verified
//
#include <hip/hip_runtime.h>
#include <math.h>
#include <stdint.h>

// Local grouped Zernike nonlinearity: bandwidth-bound 3x3 stencil over
// per-pixel channel-group soft-abs sums. Tiled into LDS via CDNA5 async
// global->LDS data mover (ASYNCcnt path); one HBM read of raw per block.

#define TILE  16
#define HALO  18               // TILE + 2 (replicate-padded halo)
#define NPIX  (HALO * HALO)    // 324
#define C_TOT 36
#define H_DIM 256
#define W_DIM 256
#define B_DIM 8

__device__ __forceinline__ void async_b128_to_lds(uint32_t lds_off, const void* gptr) {
    // GLOBAL_LOAD_ASYNC_TO_LDS_B128: vdst = LDS byte address (low 32 bits of
    // the generic shared pointer), vaddr = 64-bit global address. Tracked by
    // ASYNCcnt; EXEC-masked per lane.
    asm volatile("global_load_async_to_lds_b128 %0, %1, off"
                 :: "v"(lds_off), "v"(gptr)
                 : "memory");
}

__global__ __launch_bounds__(256) void lgz_kernel(
    const float* __restrict__ raw,
    const float* sp_b, const float* sp_a, const float* sp_m, const float* sp_e,
    const float* lo_b, const float* lo_a, const float* lo_m, const float* lo_e,
    const float* lo_g, const float* lo_p,
    const float* mi_b, const float* mi_a, const float* mi_m, const float* mi_e,
    const float* mi_g, const float* mi_p,
    const float* hi_b, const float* hi_a, const float* hi_m, const float* hi_e,
    const float* hi_g, const float* hi_p,
    float* __restrict__ out)
{
    __shared__ float rawT[NPIX][C_TOT];   // 46656 B: halo tile of raw coeffs
    __shared__ float S[NPIX][4];          //  5184 B: per-pixel group sums (slot 0 unused)

    const int tx = threadIdx.x, ty = threadIdx.y;
    const int l  = ty * TILE + tx;                  // 0..255
    const int b  = blockIdx.z;
    const int h0 = blockIdx.y * TILE;
    const int w0 = blockIdx.x * TILE;

    // Uniform scalar-parameter loads (single-element device arrays).
    const float bias[4]  = {*sp_b, *lo_b, *mi_b, *hi_b};
    const float alpha[4] = {*sp_a, *lo_a, *mi_a, *hi_a};
    const float amax[4]  = {*sp_m, *lo_m, *mi_m, *hi_m};
    const float eps[4]   = {*sp_e, *lo_e, *mi_e, *hi_e};   // eps[0] unused (special group)
    const float gss[4]   = {0.0f,  *lo_g, *mi_g, *hi_g};
    const float psat[4]  = {1.0f,  *lo_p, *mi_p, *hi_p};

    // ---- Phase 1: async-stage the 18x18 halo tile (replicate pad via clamp) ----
    const float* rawB = raw + (size_t)b * H_DIM * W_DIM * C_TOT;
#pragma unroll
    for (int it = 0; it < 2; ++it) {
        int i = l + it * 256;
        if (i < NPIX) {
            int hp = i / HALO, wp = i % HALO;
            int gh = min(max(h0 - 1 + hp, 0), H_DIM - 1);
            int gw = min(max(w0 - 1 + wp, 0), W_DIM - 1);
            const char* gp = (const char*)(rawB + ((size_t)gh * W_DIM + gw) * C_TOT);
            uint32_t lds0 = (uint32_t)(uintptr_t)(&rawT[i][0]);
#pragma unroll
            for (int k = 0; k < 9; ++k) {            // 9 x 16B = 144B per pixel
                async_b128_to_lds(lds0 + 16u * (uint32_t)k, gp + 16 * k);
            }
        }
    }
    asm volatile("s_wait_asynccnt 0x0" ::: "memory");  // my wave's async writes landed
    __syncthreads();                                   // all waves' writes visible

    // ---- Phase 2: per-pixel channel-group soft-abs sums (channels 3..35) ----
#pragma unroll
    for (int it = 0; it < 2; ++it) {
        int i = l + it * 256;
        if (i < NPIX) {
            float s1 = 0.0f, s2 = 0.0f, s3 = 0.0f;
#pragma unroll
            for (int c = 3; c < C_TOT; ++c) {
                const int g = (c < 6) ? 1 : (c < 15) ? 2 : 3;  // constant after unroll
                float u  = rawT[i][c] + bias[g];
                float sa = sqrtf(u * u + eps[g]);
                if (g == 1) s1 += sa; else if (g == 2) s2 += sa; else s3 += sa;
            }
            S[i][1] = s1; S[i][2] = s2; S[i][3] = s3;
        }
    }
    __syncthreads();

    // ---- Phase 3: 3x3 box-sum of group sums -> gains -> output ----
    const int ic = (ty + 1) * HALO + (tx + 1);  // this thread's pixel in halo coords
    float sum1 = 0.0f, sum2 = 0.0f, sum3 = 0.0f;
#pragma unroll
    for (int di = -1; di <= 1; ++di) {
#pragma unroll
        for (int dj = -1; dj <= 1; ++dj) {
            int j = ic + di * HALO + dj;
            sum1 += S[j][1]; sum2 += S[j][2]; sum3 += S[j][3];
        }
    }
    float gain[4];
    gain[0] = 1.0f;  // special group: plain affine + tanh
    gain[1] = gss[1] / (1.0f + sum1 / psat[1]);
    gain[2] = gss[2] / (1.0f + sum2 / psat[2]);
    gain[3] = gss[3] / (1.0f + sum3 / psat[3]);

    const int gh = h0 + ty, gw = w0 + tx;
    float* op = out + (((size_t)b * H_DIM + gh) * W_DIM + gw) * C_TOT;

    float res[C_TOT];
#pragma unroll
    for (int c = 0; c < C_TOT; ++c) {
        const int g = (c < 3) ? 0 : (c < 6) ? 1 : (c < 15) ? 2 : 3;
        float u = rawT[ic][c] + bias[g];
        float v = alpha[g] * gain[g] * u;
        res[c] = amax[g] * tanhf(v);
    }
#pragma unroll
    for (int k = 0; k < 9; ++k) {                 // 144B/pixel, 16B-aligned
        ((float4*)op)[k] = ((const float4*)res)[k];
    }
}

extern "C" void kernel_launch(void* const* d_in, const int* in_sizes, int n_in,
                              void* d_out, int out_size, void* d_ws, size_t ws_size,
                              hipStream_t stream) {
    (void)in_sizes; (void)n_in; (void)d_ws; (void)ws_size; (void)out_size;
    const float* raw = (const float*)d_in[0];

    dim3 grid(W_DIM / TILE, H_DIM / TILE, B_DIM);  // 16 x 16 x 8 = 2048 blocks
    dim3 block(TILE, TILE, 1);                      // 256 threads = 8 wave32

    lgz_kernel<<<grid, block, 0, stream>>>(
        raw,
        (const float*)d_in[1],  (const float*)d_in[2],  (const float*)d_in[3],  (const float*)d_in[4],
        (const float*)d_in[5],  (const float*)d_in[6],  (const float*)d_in[7],  (const float*)d_in[8],
        (const float*)d_in[9],  (const float*)d_in[10],
        (const float*)d_in[11], (const float*)d_in[12], (const float*)d_in[13], (const float*)d_in[14],
        (const float*)d_in[15], (const float*)d_in[16],
        (const float*)d_in[17], (const float*)d_in[18], (const float*)d_in[19], (const float*)d_in[20],
        (const float*)d_in[21], (const float*)d_in[22],
        (float*)d_out);
}